// TokenCompressor_9242769621264
// MI455X (gfx1250) — compile-verified
//
#include <hip/hip_runtime.h>

typedef float v2f __attribute__((ext_vector_type(2)));
typedef float v8f __attribute__((ext_vector_type(8)));
typedef int   v4i __attribute__((ext_vector_type(4)));

#define B_  4
#define T_  8192
#define D_  1024
#define C_  128
#define R_  (B_ * T_)            // 32768 token rows
#define GROUP_ROWS 64            // rows per workgroup (16 blocks of M=4)
#define HROWS 68                 // + 4 overlap rows (previous block, recomputed)
#define KC 32                    // K staging chunk
#define NSTAGE (D_ / KC)         // 32
#define LDS_AS 36                // padded LDS row stride (floats): conflict-free strided reads
#define STAGE_FLOATS (HROWS * LDS_AS)        // 2448
#define PLANE_FLOATS (GROUP_ROWS * C_)       // 8192 per plane (Ca, Za, Cb, Zb)
#define SMEM_FLOATS  (2 * STAGE_FLOATS + 4 * PLANE_FLOATS)   // 37664 fl = 150656 B
#define STAGE_ITEMS  (HROWS * (KC / 4))      // 544 float4 tiles per stage
#define NEGBIG (-1e30f)

// Async load-to-LDS path (ASYNCcnt) if the toolchain exposes the gfx1250 builtins.
#if defined(__has_builtin)
#if __has_builtin(__builtin_amdgcn_global_load_async_to_lds_b128) && \
    __has_builtin(__builtin_amdgcn_s_wait_asynccnt)
#define ASYNC_STAGE 1
#endif
#endif
#ifndef ASYNC_STAGE
#define ASYNC_STAGE 0
#endif

// Issue one K-stage of H[r0-4 .. r0+63][k0 .. k0+KC) into an LDS buffer.
// Uniform 3 iterations per thread (clamped index -> wave-uniform ASYNCcnt).
__device__ __forceinline__ void issue_stage(const float* __restrict__ H,
                                            float* __restrict__ buf,
                                            int r0, int k0, int tid)
{
    #pragma unroll
    for (int i = 0; i < 3; ++i) {
        int idx = tid + 256 * i;
        if (idx > STAGE_ITEMS - 1) idx = STAGE_ITEMS - 1;  // duplicate write, benign
        int row = idx >> 3;            // KC/4 = 8 float4 per row
        int c4  = idx & 7;
        int g   = r0 - 4 + row;
        if (g < 0) g = 0;              // garbage rows masked in epilogue
        const float* gp = H + (size_t)g * D_ + k0 + c4 * 4;
        float*       lp = buf + row * LDS_AS + c4 * 4;
#if ASYNC_STAGE
        __builtin_amdgcn_global_load_async_to_lds_b128(
            (v4i*)gp, (v4i*)lp, 0, 0);
#else
        *reinterpret_cast<float4*>(lp) = *reinterpret_cast<const float4*>(gp);
#endif
    }
}

extern "C" __global__ __launch_bounds__(256)
void tc_fused_kernel(const float* __restrict__ H,
                     const float* __restrict__ WaKV,
                     const float* __restrict__ WbKV,
                     const float* __restrict__ WaZ,
                     const float* __restrict__ WbZ,
                     const float* __restrict__ Ba,
                     const float* __restrict__ Bb,
                     float* __restrict__ Out)
{
    extern __shared__ float smem[];
    float* stage0 = smem;                          // double-buffered H staging
    float* stage1 = smem + STAGE_FLOATS;
    float* planes = smem + 2 * STAGE_FLOATS;       // 4 planes of [GROUP_ROWS][C_]

    const int tid = threadIdx.x;
    const int w   = tid >> 5;                 // wave id (wave32)
    const int l   = tid & 31;                 // lane id
    const int r0  = blockIdx.x * GROUP_ROWS;  // first "a" row of this group

    // wave -> (path, 64-col slice). Waves 0-3: a-path rows r0..r0+63.
    //                               Waves 4-7: b-path rows r0-4..r0+59 (shifted window).
    const bool isB     = (w >= 4);
    const int  cw      = (w & 3) * 64;        // 0,64,128,192 within [KV(128)|Z(128)]
    const int  colLoc  = cw & 127;            // 0 or 64 within the selected weight
    const float* Wsel  = isB ? ((cw < 128) ? WbKV : WbZ)
                             : ((cw < 128) ? WaKV : WaZ);

    const int mrow    = l & 15;               // M index within 16-row tile
    const int khalf   = (l >> 4) * 2;         // K offset 0 or 2 (A/B frag layout)
    const int rowBase = isB ? 0 : 4;          // a-path skips the 4 overlap rows

    v8f acc[4][4];
    {
        v8f zero = {0.f,0.f,0.f,0.f,0.f,0.f,0.f,0.f};
        #pragma unroll
        for (int t = 0; t < 4; ++t)
            #pragma unroll
            for (int ct = 0; ct < 4; ++ct)
                acc[t][ct] = zero;
    }

    // Software pipeline: prefetch stage 0, then per stage:
    //   wait(own stage-k loads) -> barrier -> issue stage k+1 -> compute stage k.
    issue_stage(H, stage0, r0, 0, tid);

    for (int kt = 0; kt < NSTAGE; ++kt) {
        float* cur = (kt & 1) ? stage1 : stage0;
        float* nxt = (kt & 1) ? stage0 : stage1;

#if ASYNC_STAGE
        __builtin_amdgcn_s_wait_asynccnt(0);  // my stage-kt writes are in LDS
#endif
        __syncthreads();                      // all waves ready; prev readers of nxt done

        if (kt + 1 < NSTAGE)
            issue_stage(H, nxt, r0, (kt + 1) * KC, tid);  // overlaps with compute below

        const int k0 = kt * KC;
        #pragma unroll
        for (int kk = 0; kk < KC; kk += 4) {
            // A fragments: 16x4 f32, lane M = mrow, (K,K+1) pair -> ds_load_b64
            v2f afr[4];
            #pragma unroll
            for (int t = 0; t < 4; ++t) {
                const float* ap = cur + (rowBase + 16 * t + mrow) * LDS_AS + kk + khalf;
                afr[t] = *reinterpret_cast<const v2f*>(ap);
            }
            #pragma unroll
            for (int ct = 0; ct < 4; ++ct) {
                // B fragment: 4x16 f32, lane N = mrow, VGPR0/1 = W rows k, k+1
                const int col  = colLoc + 16 * ct + mrow;
                const int krow = k0 + kk + khalf;
                v2f bfr;
                bfr.x = Wsel[(size_t)krow * C_ + col];
                bfr.y = Wsel[(size_t)(krow + 1) * C_ + col];
                #pragma unroll
                for (int t = 0; t < 4; ++t) {
                    acc[t][ct] = __builtin_amdgcn_wmma_f32_16x16x4_f32(
                        false, afr[t], false, bfr,
                        (short)0, acc[t][ct], false, false);
                }
            }
        }
    }

    __syncthreads();
    // Scatter accumulators to LDS planes: C/D layout -> row = 16t + v + 8*(l>=16), col = l&15
    {
        const int p = (isB ? 2 : 0) + ((cw < 128) ? 0 : 1);   // 0:Ca 1:Za 2:Cb 3:Zb
        float* pl = planes + p * PLANE_FLOATS;
        const int rhalf = (l >> 4) * 8;
        #pragma unroll
        for (int t = 0; t < 4; ++t)
            #pragma unroll
            for (int ct = 0; ct < 4; ++ct)
                #pragma unroll
                for (int v = 0; v < 8; ++v) {
                    int row = 16 * t + v + rhalf;
                    int col = colLoc + 16 * ct + mrow;
                    pl[row * C_ + col] = acc[t][ct][v];
                }
    }
    __syncthreads();

    // Epilogue: per (block, channel) softmax over 8-entry window + weighted sum.
    const float* Ca = planes + 0 * PLANE_FLOATS;
    const float* Za = planes + 1 * PLANE_FLOATS;
    const float* Cb = planes + 2 * PLANE_FLOATS;  // row i == global row r0-4+i
    const float* Zb = planes + 3 * PLANE_FLOATS;
    const bool batchStart = ((r0 % T_) == 0);

    for (int item = tid; item < (GROUP_ROWS / 4) * C_; item += 256) {
        int blk = item >> 7;                  // 0..15 local block
        int n   = item & 127;                 // channel
        bool mask = batchStart && (blk == 0); // first block of a batch: no previous block
        float z[8], vv[8];
        #pragma unroll
        for (int j = 0; j < 4; ++j) {
            int rr = 4 * blk + j;
            z[j]      = Za[rr * C_ + n] + Ba[j * C_ + n];
            vv[j]     = Ca[rr * C_ + n];
            z[4 + j]  = mask ? NEGBIG : (Zb[rr * C_ + n] + Bb[j * C_ + n]);
            vv[4 + j] = mask ? 0.f    :  Cb[rr * C_ + n];
        }
        float m = z[0];
        #pragma unroll
        for (int j = 1; j < 8; ++j) m = fmaxf(m, z[j]);
        float s = 0.f, o = 0.f;
        #pragma unroll
        for (int j = 0; j < 8; ++j) {
            float e = expf(z[j] - m);
            s += e;
            o += e * vv[j];
        }
        int gb = (r0 >> 2) + blk;             // global block index
        Out[(size_t)gb * C_ + n] = o / s;
    }
}

extern "C" void kernel_launch(void* const* d_in, const int* in_sizes, int n_in,
                              void* d_out, int out_size, void* d_ws, size_t ws_size,
                              hipStream_t stream) {
    (void)in_sizes; (void)n_in; (void)out_size; (void)d_ws; (void)ws_size;
    const float* H    = (const float*)d_in[0];
    const float* WaKV = (const float*)d_in[1];
    const float* WbKV = (const float*)d_in[2];
    const float* WaZ  = (const float*)d_in[3];
    const float* WbZ  = (const float*)d_in[4];
    const float* Ba   = (const float*)d_in[5];
    const float* Bb   = (const float*)d_in[6];
    float* Out = (float*)d_out;

    const int smemBytes = SMEM_FLOATS * (int)sizeof(float);   // ~150.7 KB
    (void)hipFuncSetAttribute(reinterpret_cast<const void*>(tc_fused_kernel),
                              hipFuncAttributeMaxDynamicSharedMemorySize, smemBytes);

    dim3 grid(R_ / GROUP_ROWS);   // 512 workgroups
    dim3 block(256);              // 8 wave32 waves
    tc_fused_kernel<<<grid, block, smemBytes, stream>>>(
        H, WaKV, WbKV, WaZ, WbZ, Ba, Bb, Out);
}